// MMDNE_31851477467218
// MI455X (gfx1250) — compile-verified
//
#include <hip/hip_runtime.h>

typedef __attribute__((ext_vector_type(2))) float v2f;
typedef __attribute__((ext_vector_type(8))) float v8f;

namespace {
constexpr int kB = 50000;
constexpr int kF = 128;
constexpr int kE = 32;
constexpr int kH = 2;
constexpr int kN = 10;
constexpr int kWavesTotal = kB / 16;          // 3125 waves, 16 batch elements each
constexpr int kWavesPerBlock = 2;
constexpr int kThreads = 32 * kWavesPerBlock; // wave32
constexpr int EST = kE + 1;                   // padded LDS row stride (33 floats)
constexpr int kKSteps = kF / 4;               // 32 k-steps of K=4
constexpr int WFS = 48;                       // Wfrag row-pair stride in float2 (96 floats
                                              // == 32 mod 64 banks -> halves hit disjoint banks)
}

// One 16-row M-tile of the embedding GEMM: D(16x32) = A(16x128) * W(128x32) + b
// using V_WMMA_F32_16X16X4_F32 (exact fp32), two 16x16 N-tiles, 32 k-steps.
// rowBase: per-lane pointer to source row (row M = lane&15).
// wf: fragment-swizzled W in LDS: wf[(kk*2+rp)*WFS + c] = {W[4kk+2rp][c], W[4kk+2rp+1][c]}
//     -> each B fragment is one aligned, bank-conflict-free ds_load_b64.
__device__ __forceinline__ void gemm16_tile(
    const float* __restrict__ rowBase,
    const v2f* __restrict__ wf,
    const float* __restrict__ sBias,  // [32] in LDS
    float* __restrict__ dst,          // LDS tile, row stride EST
    int half, int lm)
{
    v8f c0 = {}; v8f c1 = {};
    const float* rp = rowBase + 2 * half;   // lanes 16-31 supply K = k+2, k+3
    #pragma unroll 4
    for (int kk = 0; kk < kKSteps; ++kk) {
        // A fragment: VGPR0 = K 4kk+2*half, VGPR1 = K 4kk+1+2*half (ISA 16x4 f32 layout)
        v2f a = *(const v2f*)(rp + 4 * kk);
        const v2f* wrow = wf + (kk * 2 + half) * WFS;
        v2f b0 = wrow[lm];        // N-tile 0: cols 0..15
        v2f b1 = wrow[16 + lm];   // N-tile 1: cols 16..31
        c0 = __builtin_amdgcn_wmma_f32_16x16x4_f32(false, a, false, b0, (short)0, c0, false, false);
        c1 = __builtin_amdgcn_wmma_f32_16x16x4_f32(false, a, false, b1, (short)0, c1, false, false);
    }
    // C/D layout: VGPR j -> row (j + 8*half), column lane&15 (c0) / 16+(lane&15) (c1)
    const float bias0 = sBias[lm];
    const float bias1 = sBias[16 + lm];
    #pragma unroll
    for (int j = 0; j < 8; ++j) {
        const int r = j + 8 * half;
        dst[r * EST + lm]      = c0[j] + bias0;
        dst[r * EST + 16 + lm] = c1[j] + bias1;
    }
}

__global__ __launch_bounds__(kThreads) void mmdne_fused_kernel(
    const float* __restrict__ s_fts, const float* __restrict__ t_fts,
    const float* __restrict__ s_h_fts, const float* __restrict__ t_h_fts,
    const float* __restrict__ s_neg_fts, const float* __restrict__ t_neg_fts,
    const float* __restrict__ event_time,
    const float* __restrict__ s_h_times, const float* __restrict__ t_h_times,
    const float* __restrict__ s_h_mask, const float* __restrict__ t_h_mask,
    const float* __restrict__ W_fts, const float* __restrict__ b_fts,
    const float* __restrict__ a_vec,
    const float* __restrict__ delta_s_p, const float* __restrict__ delta_t_p,
    float* __restrict__ out)
{
    __shared__ v2f   sWf[kKSteps * 2 * WFS];             // 24 KB fragment-swizzled W
    __shared__ float sA[2 * kE];
    __shared__ float sBias[kE];
    __shared__ float sEmb[kWavesPerBlock][6 * 16 * EST]; // s,t,s_h0,s_h1,t_h0,t_h1
    __shared__ float sNeg[kWavesPerBlock][2 * 16 * EST]; // t_neg tile, s_neg tile

    const int tid = threadIdx.x;
    // Build fragment-native W layout (each global element read exactly once, coalesced)
    for (int idx = tid; idx < kKSteps * 2 * kE; idx += kThreads) {
        const int kk = idx >> 6;
        const int rp = (idx >> 5) & 1;
        const int c  = idx & 31;
        const int row = 4 * kk + 2 * rp;
        v2f p;
        p.x = W_fts[row * kE + c];
        p.y = W_fts[(row + 1) * kE + c];
        sWf[(kk * 2 + rp) * WFS + c] = p;
    }
    if (tid < 2 * kE) sA[tid] = a_vec[tid];
    if (tid < kE) sBias[tid] = b_fts[tid];
    __syncthreads();

    const int wv   = tid >> 5;
    const int lane = tid & 31;
    const int half = lane >> 4;   // 0: s-side lanes, 1: t-side lanes
    const int lm   = lane & 15;

    const int  wgRaw = blockIdx.x * kWavesPerBlock + wv;
    const bool valid = wgRaw < kWavesTotal;
    const int  wg    = valid ? wgRaw : (kWavesTotal - 1);  // clamp: uniform flow for barriers/WMMA
    const long long e = (long long)wg * 16 + lm;           // this lane's batch element

    float* emb = sEmb[wv];
    float* neg = sNeg[wv];

    // ---- Phase 1: positive embeddings via WMMA ----
    gemm16_tile(s_fts + e * kF,              sWf, sBias, emb + 0 * 16 * EST, half, lm);
    gemm16_tile(t_fts + e * kF,              sWf, sBias, emb + 1 * 16 * EST, half, lm);
    gemm16_tile(s_h_fts + (e * kH + 0) * kF, sWf, sBias, emb + 2 * 16 * EST, half, lm);
    gemm16_tile(s_h_fts + (e * kH + 1) * kF, sWf, sBias, emb + 3 * 16 * EST, half, lm);
    gemm16_tile(t_h_fts + (e * kH + 0) * kF, sWf, sBias, emb + 4 * 16 * EST, half, lm);
    gemm16_tile(t_h_fts + (e * kH + 1) * kF, sWf, sBias, emb + 5 * 16 * EST, half, lm);
    __syncthreads();

    // ---- Phase 2: attention weights + p_lambda ----
    const float* cur = emb + (half ? 1 : 0) * 16 * EST + lm * EST;       // s_emb | t_emb
    const float* oth = emb + (half ? 0 : 1) * 16 * EST + lm * EST;       // the other
    const float* h0  = emb + (2 + 2 * half + 0) * 16 * EST + lm * EST;   // {s,t}_h_emb[0]
    const float* h1  = emb + (2 + 2 * half + 1) * 16 * EST + lm * EST;   // {s,t}_h_emb[1]

    const float  ev     = event_time[e];
    const float* htimes = half ? t_h_times : s_h_times;
    const float* hmask  = half ? t_h_mask  : s_h_mask;
    const float  delta  = half ? delta_t_p[0] : delta_s_p[0];

    float sc_cur = 0.f, sc_h0 = 0.f, sc_h1 = 0.f;
    float pmu = 0.f, pa0 = 0.f, pa1 = 0.f;
    #pragma unroll
    for (int k = 0; k < kE; ++k) {
        const float c = cur[k], o = oth[k], x0 = h0[k], x1 = h1[k];
        sc_cur += c  * sA[k];
        sc_h0  += x0 * sA[kE + k];
        sc_h1  += x1 * sA[kE + k];
        const float d  = c  - o; pmu += d  * d;
        const float d0 = x0 - o; pa0 += d0 * d0;
        const float d1 = x1 - o; pa1 += d1 * d1;
    }
    const float dt0 = fabsf(ev - htimes[e * kH + 0]);
    const float dt1 = fabsf(ev - htimes[e * kH + 1]);
    float sim0 = __expf(-delta * dt0) * (sc_cur + sc_h0);
    float sim1 = __expf(-delta * dt1) * (sc_cur + sc_h1);
    sim0 = sim0 >= 0.f ? sim0 : 0.2f * sim0;   // leaky_relu(0.2)
    sim1 = sim1 >= 0.f ? sim1 : 0.2f * sim1;
    const float mx  = fmaxf(sim0, sim1);
    const float e0  = __expf(sim0 - mx), e1 = __expf(sim1 - mx);
    const float inv = 1.f / (e0 + e1);
    const float w0  = e0 * inv * __expf(delta * dt0) * hmask[e * kH + 0];
    const float w1  = e1 * inv * __expf(delta * dt1) * hmask[e * kH + 1];

    // p_lambda = p_mu + sum_h w_s[h]*p_alpha_s[h]  (s-side lanes own it)
    const float plam = -pmu - (w0 * pa0 + w1 * pa1);
    if (valid && half == 0) out[e] = plam;

    // ---- Phase 3: stream negatives through WMMA, consume immediately ----
    for (int n = 0; n < kN; ++n) {
        // half==0 lanes need t_neg embeddings (n_lambda_s); half==1 need s_neg (n_lambda_t)
        gemm16_tile(t_neg_fts + (e * kN + n) * kF, sWf, sBias, neg + 0 * 16 * EST, half, lm);
        gemm16_tile(s_neg_fts + (e * kN + n) * kF, sWf, sBias, neg + 1 * 16 * EST, half, lm);
        __syncthreads();

        const float* nb = neg + half * 16 * EST + lm * EST;
        float mu = 0.f, a0 = 0.f, a1 = 0.f;
        #pragma unroll
        for (int k = 0; k < kE; ++k) {
            const float y  = nb[k];
            const float d  = cur[k] - y; mu += d  * d;
            const float d0 = h0[k]  - y; a0 += d0 * d0;
            const float d1 = h1[k]  - y; a1 += d1 * d1;
        }
        const float lam = -mu - (w0 * a0 + w1 * a1);
        if (valid) out[(long long)kB + (long long)half * kB * kN + e * kN + n] = lam;
        __syncthreads();   // WAR guard before next tile overwrites sNeg
    }
}

extern "C" void kernel_launch(void* const* d_in, const int* in_sizes, int n_in,
                              void* d_out, int out_size, void* d_ws, size_t ws_size,
                              hipStream_t stream) {
    (void)in_sizes; (void)n_in; (void)out_size; (void)d_ws; (void)ws_size;
    const float* s_fts      = (const float*)d_in[0];
    const float* t_fts      = (const float*)d_in[1];
    const float* s_h_fts    = (const float*)d_in[2];
    const float* t_h_fts    = (const float*)d_in[3];
    const float* s_neg_fts  = (const float*)d_in[4];
    const float* t_neg_fts  = (const float*)d_in[5];
    const float* event_time = (const float*)d_in[6];
    const float* s_h_times  = (const float*)d_in[7];
    const float* t_h_times  = (const float*)d_in[8];
    const float* s_h_mask   = (const float*)d_in[9];
    const float* t_h_mask   = (const float*)d_in[10];
    const float* W_fts      = (const float*)d_in[11];
    const float* b_fts      = (const float*)d_in[12];
    const float* a_vec      = (const float*)d_in[13];
    const float* delta_s    = (const float*)d_in[14];
    const float* delta_t    = (const float*)d_in[15];
    float* out = (float*)d_out;

    const int blocks = (kWavesTotal + kWavesPerBlock - 1) / kWavesPerBlock; // 1563
    mmdne_fused_kernel<<<blocks, kThreads, 0, stream>>>(
        s_fts, t_fts, s_h_fts, t_h_fts, s_neg_fts, t_neg_fts,
        event_time, s_h_times, t_h_times, s_h_mask, t_h_mask,
        W_fts, b_fts, a_vec, delta_s, delta_t, out);
}